// MORNLayer_54709293416891
// MI455X (gfx1250) — compile-verified
//
#include <hip/hip_runtime.h>
#include <hip/hip_bf16.h>
#include <cstddef>

// ---------------------------------------------------------------------------
// HGT layer for MI455X (gfx1250, wave32, WMMA).
// Pipeline per relation r (src type s, dst type d):
//   1. fold:  Wk' = Wk[s] @ blockdiag(rel_att[r]),  Wv' = Wv[s] @ blockdiag(rel_msg[r])
//   2. gemm:  k = x_s @ Wk' + bk',  v = x_s @ Wv' + bv',  q = x_d @ Wq[d] + bq[d]
//   3. edge passes: den -> logits(+atomicMax) -> exp(+atomicAdd) -> msg scatter
// Then per type: trans = agg @ Wa[t] + ba[t]; skip-mix; LayerNorm -> out.
// ---------------------------------------------------------------------------

typedef __attribute__((ext_vector_type(2))) float v2f;
typedef __attribute__((ext_vector_type(8))) float v8f;

#define HGT_EPS 1e-9f
#define HGT_LNEPS 1e-5f

// ------------------------- generic fill -------------------------------------
__global__ void hgt_fill(float* __restrict__ p, float v, int n) {
  int i = blockIdx.x * blockDim.x + threadIdx.x;
  if (i < n) p[i] = v;
}

// ------------------- fold rel matrix into W (and bias) ----------------------
// Wout[i, h*16+e] = sum_d W[i, h*16+d] * R[h, d, e];   i==128 row folds bias.
__global__ void hgt_fold(const float* __restrict__ W, const float* __restrict__ b,
                         const float* __restrict__ R,
                         float* __restrict__ Wout, float* __restrict__ bout) {
  int idx = blockIdx.x * blockDim.x + threadIdx.x;
  if (idx >= 129 * 128) return;
  int i = idx >> 7;
  int j = idx & 127;
  int h = j >> 4, e = j & 15;
  const float* Rh = R + h * 256;
  float acc = 0.f;
  if (i < 128) {
    const float* Wr = W + i * 128 + h * 16;
#pragma unroll
    for (int d = 0; d < 16; ++d) acc += Wr[d] * Rh[d * 16 + e];
    Wout[i * 128 + j] = acc;
  } else {
    const float* br = b + h * 16;
#pragma unroll
    for (int d = 0; d < 16; ++d) acc += br[d] * Rh[d * 16 + e];
    bout[j] = acc;
  }
}

// ------------------------- WMMA f32 GEMM ------------------------------------
// C[M x 128] = A[M x 128] @ B[128 x 128] + bias.  One block = 8 waves = 128
// rows of C; each wave owns a 16-row strip and 8 16x16 tiles across N.
// B is staged TRANSPOSED in LDS with an XOR-swizzled k index:
//   BsT[col*128 + (k ^ 4*(col&15))] = B[k][col]
// so each B fragment (k,k+1 for one column) is a single aligned ds_load_b64,
// and the 16 lanes of a half-wave hit 16 distinct LDS banks.
__global__ __launch_bounds__(256) void hgt_gemm128(
    const float* __restrict__ A, const float* __restrict__ B,
    const float* __restrict__ bias, float* __restrict__ C, int M) {
  __shared__ float BsT[128 * 128];  // 64 KB
  for (int i = threadIdx.x; i < 128 * 128; i += 256) {
    int kk = i >> 7;          // source row (k)
    int cc = i & 127;         // source col (n)
    BsT[cc * 128 + (kk ^ ((cc & 15) * 4))] = B[i];
  }
  __syncthreads();

  const int wave = threadIdx.x >> 5;
  const int lane = threadIdx.x & 31;
  const int mlo  = lane & 15;
  const int sel  = lane >> 4;        // which K half-pair this lane holds
  const int swz  = mlo * 4;          // XOR swizzle for this lane's column
  const int rowBase = blockIdx.x * 128 + wave * 16;

  int arowi = rowBase + mlo;
  if (arowi >= M) arowi = M - 1;     // clamp loads; stores are guarded
  const float* arow = A + (size_t)arowi * 128;

  // prefetch this lane's full A row (512 B) ahead of the K loop
  __builtin_prefetch(arow, 0, 1);
  __builtin_prefetch(arow + 64, 0, 1);

  v8f acc[8];
#pragma unroll
  for (int nt = 0; nt < 8; ++nt) {
    float b0 = bias[nt * 16 + mlo];  // same column for all 8 rows of the tile
    acc[nt] = (v8f){b0, b0, b0, b0, b0, b0, b0, b0};
  }

#pragma unroll 4
  for (int k = 0; k < 128; k += 4) {
    const int ka = k + sel * 2;
    v2f a = *(const v2f*)&arow[ka];                 // global_load_b64
    const int kswz = ka ^ swz;                      // even -> 8B aligned
#pragma unroll
    for (int nt = 0; nt < 8; ++nt) {
      v2f b = *(const v2f*)&BsT[nt * 2048 + mlo * 128 + kswz];  // ds_load_b64
      acc[nt] = __builtin_amdgcn_wmma_f32_16x16x4_f32(
          false, a, false, b, (short)0, acc[nt], false, false);
    }
  }

#pragma unroll
  for (int nt = 0; nt < 8; ++nt) {
#pragma unroll
    for (int j = 0; j < 8; ++j) {
      int r = rowBase + sel * 8 + j;  // C layout: vgpr j -> row j + sel*8
      if (r < M) C[(size_t)r * 128 + nt * 16 + mlo] = acc[nt][j];
    }
  }
}

// --------------------- edge pass 0: weight denominator ----------------------
__global__ void hgt_edge_den(const float* __restrict__ ew, const int* __restrict__ dst,
                             float* __restrict__ den, int E) {
  int e = blockIdx.x * blockDim.x + threadIdx.x;
  if (e < E) atomicAdd(&den[dst[e]], fmaxf(ew[e], 0.f));
}

// --------------------- edge pass 1: logits + segment max --------------------
__global__ void hgt_edge_logits(const float* __restrict__ q, const float* __restrict__ k,
                                const int* __restrict__ src, const int* __restrict__ dst,
                                const float* __restrict__ ew, const float* __restrict__ den,
                                const float* __restrict__ rpri,
                                float* __restrict__ t, float* __restrict__ m, int E) {
  int idx = blockIdx.x * blockDim.x + threadIdx.x;
  if (idx >= E * 8) return;
  int e = idx >> 3, h = idx & 7;
  int s = src[e], d = dst[e];
  const float* qp = q + (size_t)d * 128 + h * 16;
  const float* kp = k + (size_t)s * 128 + h * 16;
  float acc = 0.f;
#pragma unroll
  for (int i = 0; i < 16; ++i) acc += qp[i] * kp[i];
  float w   = fmaxf(ew[e], 0.f);
  float ewn = w / fmaxf(den[d], HGT_EPS);
  float val = (acc + __logf(ewn + HGT_EPS)) * rpri[h] * 0.25f;  // 1/sqrt(DK=16)
  t[idx] = val;
  atomicMax(&m[(size_t)d * 8 + h], val);     // global_atomic_max_num_f32
}

// --------------------- edge pass 2: exp + segment sum -----------------------
__global__ void hgt_edge_exp(float* __restrict__ t, const int* __restrict__ dst,
                             const float* __restrict__ m, float* __restrict__ s, int E) {
  int idx = blockIdx.x * blockDim.x + threadIdx.x;
  if (idx >= E * 8) return;
  int e = idx >> 3, h = idx & 7;
  int d = dst[e];
  float p = __expf(t[idx] - m[(size_t)d * 8 + h]);
  t[idx] = p;
  atomicAdd(&s[(size_t)d * 8 + h], p);
}

// --------------------- edge pass 3: weighted message scatter ----------------
__global__ void hgt_edge_msg(const float* __restrict__ v, const float* __restrict__ t,
                             const int* __restrict__ src, const int* __restrict__ dst,
                             const float* __restrict__ s, float* __restrict__ agg, int E) {
  int idx = blockIdx.x * blockDim.x + threadIdx.x;
  if (idx >= E * 8) return;
  int e = idx >> 3, h = idx & 7;
  int sn = src[e], d = dst[e];
  float a = t[idx] / s[(size_t)d * 8 + h];
  const float* vp = v + (size_t)sn * 128 + h * 16;
  float* op = agg + (size_t)d * 128 + h * 16;
#pragma unroll
  for (int i = 0; i < 16; ++i) atomicAdd(op + i, vp[i] * a);
}

// --------------------- skip-mix + LayerNorm (wave32 per row) ----------------
__global__ __launch_bounds__(256) void hgt_combine_ln(
    const float* __restrict__ trans, const float* __restrict__ base,
    const float* __restrict__ g, const float* __restrict__ bvec,
    const float* __restrict__ skip, int nid, float* __restrict__ out, int N) {
  int row  = blockIdx.x * 8 + (threadIdx.x >> 5);
  int lane = threadIdx.x & 31;
  if (row >= N) return;
  float alpha = 1.f / (1.f + __expf(-skip[nid]));
  float4 t4 = ((const float4*)(trans + (size_t)row * 128))[lane];
  float4 b4 = ((const float4*)(base + (size_t)row * 128))[lane];
  float4 o;
  o.x = t4.x * alpha + b4.x * (1.f - alpha);
  o.y = t4.y * alpha + b4.y * (1.f - alpha);
  o.z = t4.z * alpha + b4.z * (1.f - alpha);
  o.w = t4.w * alpha + b4.w * (1.f - alpha);
  float ssum = o.x + o.y + o.z + o.w;
#pragma unroll
  for (int mk = 16; mk >= 1; mk >>= 1) ssum += __shfl_xor(ssum, mk, 32);
  float mu = ssum * (1.f / 128.f);
  float dx = o.x - mu, dy = o.y - mu, dz = o.z - mu, dw = o.w - mu;
  float vsum = dx * dx + dy * dy + dz * dz + dw * dw;
#pragma unroll
  for (int mk = 16; mk >= 1; mk >>= 1) vsum += __shfl_xor(vsum, mk, 32);
  float inv = rsqrtf(vsum * (1.f / 128.f) + HGT_LNEPS);
  float4 g4  = ((const float4*)g)[lane];
  float4 bb4 = ((const float4*)bvec)[lane];
  float4 r;
  r.x = dx * inv * g4.x + bb4.x;
  r.y = dy * inv * g4.y + bb4.y;
  r.z = dz * inv * g4.z + bb4.z;
  r.w = dw * inv * g4.w + bb4.w;
  ((float4*)(out + (size_t)row * 128))[lane] = r;
}

// ---------------------------------------------------------------------------
static void run_relation(const float* x_s, int Ns, const float* x_d, int Nd,
                         const int* src, const int* dst, const float* ew, int E,
                         const float* Wk_t, const float* bk_t,
                         const float* Wq_t, const float* bq_t,
                         const float* Wv_t, const float* bv_t,
                         const float* ratt, const float* rmsg, const float* rpri,
                         float* kbuf, float* vbuf, float* qbuf, float* agg,
                         float* tbuf, float* mbuf, float* sbuf, float* den,
                         float* Wf, float* bf, hipStream_t stream) {
  const int T = 256;
  hgt_fold<<<(129 * 128 + T - 1) / T, T, 0, stream>>>(Wk_t, bk_t, ratt, Wf, bf);
  hgt_fold<<<(129 * 128 + T - 1) / T, T, 0, stream>>>(Wv_t, bv_t, rmsg, Wf + 16384, bf + 128);
  hgt_gemm128<<<(Ns + 127) / 128, T, 0, stream>>>(x_s, Wf, bf, kbuf, Ns);
  hgt_gemm128<<<(Ns + 127) / 128, T, 0, stream>>>(x_s, Wf + 16384, bf + 128, vbuf, Ns);
  hgt_gemm128<<<(Nd + 127) / 128, T, 0, stream>>>(x_d, Wq_t, bq_t, qbuf, Nd);

  hgt_fill<<<(Nd + T - 1) / T, T, 0, stream>>>(den, 0.f, Nd);
  hgt_fill<<<(Nd * 8 + T - 1) / T, T, 0, stream>>>(mbuf, -3.0e38f, Nd * 8);
  hgt_fill<<<(Nd * 8 + T - 1) / T, T, 0, stream>>>(sbuf, 0.f, Nd * 8);
  hgt_fill<<<(Nd * 128 + T - 1) / T, T, 0, stream>>>(agg, 0.f, Nd * 128);

  hgt_edge_den<<<(E + T - 1) / T, T, 0, stream>>>(ew, dst, den, E);
  hgt_edge_logits<<<(E * 8 + T - 1) / T, T, 0, stream>>>(qbuf, kbuf, src, dst, ew, den,
                                                         rpri, tbuf, mbuf, E);
  hgt_edge_exp<<<(E * 8 + T - 1) / T, T, 0, stream>>>(tbuf, dst, mbuf, sbuf, E);
  hgt_edge_msg<<<(E * 8 + T - 1) / T, T, 0, stream>>>(vbuf, tbuf, src, dst, sbuf, agg, E);
}

extern "C" void kernel_launch(void* const* d_in, const int* in_sizes, int n_in,
                              void* d_out, int out_size, void* d_ws, size_t ws_size,
                              hipStream_t stream) {
  const float* x_a  = (const float*)d_in[0];
  const float* x_b  = (const float*)d_in[1];
  const float* ew0  = (const float*)d_in[2];
  const float* ew1  = (const float*)d_in[3];
  const int*   src0 = (const int*)d_in[4];
  const int*   dst0 = (const int*)d_in[5];
  const int*   src1 = (const int*)d_in[6];
  const int*   dst1 = (const int*)d_in[7];
  const float* Wk   = (const float*)d_in[8];
  const float* bk   = (const float*)d_in[9];
  const float* Wq   = (const float*)d_in[10];
  const float* bq   = (const float*)d_in[11];
  const float* Wv   = (const float*)d_in[12];
  const float* bv   = (const float*)d_in[13];
  const float* Wa   = (const float*)d_in[14];
  const float* ba   = (const float*)d_in[15];
  const float* ln_g = (const float*)d_in[16];
  const float* ln_b = (const float*)d_in[17];
  const float* rpri = (const float*)d_in[18];
  const float* ratt = (const float*)d_in[19];
  const float* rmsg = (const float*)d_in[20];
  const float* skip = (const float*)d_in[21];

  const int NA = in_sizes[0] / 128;
  const int NB = in_sizes[1] / 128;
  const int E0 = in_sizes[2];
  const int E1 = in_sizes[3];
  const int NM = NA > NB ? NA : NB;
  const int EM = E0 > E1 ? E0 : E1;

  float* ws = (float*)d_ws;
  size_t o = 0;
  float* kbuf  = ws + o; o += (size_t)NM * 128;
  float* vbuf  = ws + o; o += (size_t)NM * 128;
  float* qbuf  = ws + o; o += (size_t)NM * 128;
  float* agg_a = ws + o; o += (size_t)NA * 128;
  float* agg_b = ws + o; o += (size_t)NB * 128;
  float* tbuf  = ws + o; o += (size_t)EM * 8;
  float* mbuf  = ws + o; o += (size_t)NM * 8;
  float* sbuf  = ws + o; o += (size_t)NM * 8;
  float* den   = ws + o; o += (size_t)NM;
  float* Wf    = ws + o; o += 2 * 16384;
  float* bf    = ws + o; o += 2 * 128;
  (void)o; (void)ws_size; (void)n_in; (void)out_size;

  // relation 0: a (type 0) -> b (type 1)
  run_relation(x_a, NA, x_b, NB, src0, dst0, ew0, E0,
               Wk + 0 * 16384, bk + 0 * 128, Wq + 1 * 16384, bq + 1 * 128,
               Wv + 0 * 16384, bv + 0 * 128, ratt + 0 * 2048, rmsg + 0 * 2048,
               rpri + 0 * 8, kbuf, vbuf, qbuf, agg_b, tbuf, mbuf, sbuf, den,
               Wf, bf, stream);
  // relation 1: b (type 1) -> a (type 0)
  run_relation(x_b, NB, x_a, NA, src1, dst1, ew1, E1,
               Wk + 1 * 16384, bk + 1 * 128, Wq + 0 * 16384, bq + 0 * 128,
               Wv + 1 * 16384, bv + 1 * 128, ratt + 1 * 2048, rmsg + 1 * 2048,
               rpri + 1 * 8, kbuf, vbuf, qbuf, agg_a, tbuf, mbuf, sbuf, den,
               Wf, bf, stream);

  float* out   = (float*)d_out;
  float* trans = kbuf;  // k/v/q buffers are dead after edge passes; reuse
  const int T = 256;

  // type 0 (a)
  hgt_gemm128<<<(NA + 127) / 128, T, 0, stream>>>(agg_a, Wa + 0 * 16384, ba + 0 * 128,
                                                  trans, NA);
  hgt_combine_ln<<<(NA + 7) / 8, T, 0, stream>>>(trans, x_a, ln_g + 0 * 128,
                                                 ln_b + 0 * 128, skip, 0, out, NA);
  // type 1 (b)
  hgt_gemm128<<<(NB + 127) / 128, T, 0, stream>>>(agg_b, Wa + 1 * 16384, ba + 1 * 128,
                                                  trans, NB);
  hgt_combine_ln<<<(NB + 7) / 8, T, 0, stream>>>(trans, x_b, ln_g + 1 * 128,
                                                 ln_b + 1 * 128, skip, 1,
                                                 out + (size_t)NA * 128, NB);
}